// GATConv_4466765988375
// MI455X (gfx1250) — compile-verified
//
#include <hip/hip_runtime.h>

#define NN 50000
#define NE 1600000
#define EF 16

typedef __attribute__((ext_vector_type(2))) float v2f;
typedef __attribute__((ext_vector_type(8))) float v8f;

__device__ __forceinline__ float leaky(float x) { return x > 0.f ? x : 0.2f * x; }

// Mixed-sign float atomic max (dst initialized to -inf). Order-independent result.
__device__ __forceinline__ void atomicMaxF(float* addr, float v) {
    if (v >= 0.f) atomicMax((int*)addr, __float_as_int(v));
    else          atomicMin((unsigned int*)addr, __float_as_uint(v));
}

__global__ void k_fill(float* __restrict__ p, float v, int n) {
    int i = blockIdx.x * blockDim.x + threadIdx.x;
    if (i < n) p[i] = v;
}

// ---- edge encoder: emb = relu(e_feat @ We + be), scatter-add to dst ----
// e_feat is a 102MB one-shot stream: non-temporal loads keep it out of the
// L2 working set (z1/out1 must stay resident for the gather/scatter passes).
__global__ void k_edge_embed(const float* __restrict__ ef, const float* __restrict__ We,
                             const float* __restrict__ be, const int* __restrict__ dst,
                             float* __restrict__ n_e) {
    int e = blockIdx.x * blockDim.x + threadIdx.x;
    if (e >= NE) return;
    float acc = be[0];
#pragma unroll
    for (int j = 0; j < EF; ++j)
        acc += __builtin_nontemporal_load(ef + (size_t)e * EF + j) * We[j];
    acc = fmaxf(acc, 0.f);
    atomicAdd(&n_e[dst[e]], acc);
}

// ---- feats = concat(in_feat[N,127], n_e[N,1]) ----
__global__ void k_feats(const float* __restrict__ in_feat, const float* __restrict__ n_e,
                        float* __restrict__ feats) {
    int g = blockIdx.x * blockDim.x + threadIdx.x;
    if (g >= NN * 128) return;
    int n = g >> 7, c = g & 127;
    feats[g] = (c < 127) ? in_feat[n * 127 + c] : n_e[n];
}

// ---- fp32 WMMA GEMM: C[M,NCOL] = A[M,K] @ B[K,NCOL] ----
// One wave computes a 16 x (16*NT) strip: A fragment reused NT times.
// K, NCOL compile-time -> all loads are immediate offsets off one base reg.
// A 16x4 f32 fragment: lanes 0-15 M=lane (K=k,k+1), lanes 16-31 M=lane-16 (K=k+2,k+3)
// B 4x16 f32 fragment: same K split, N = lane&15
// C 16x16 f32: VGPR v -> M = v + (lane<16 ? 0 : 8), N = lane&15
template <int K, int NCOL, int NT>
__global__ void k_gemm_t(const float* __restrict__ A, const float* __restrict__ B,
                         float* __restrict__ C, int M) {
    constexpr int TILEW = 16 * NT;
    constexpr int NGRP  = (NCOL + TILEW - 1) / TILEW;
    int gwave = (blockIdx.x * blockDim.x + threadIdx.x) >> 5;
    int lane  = threadIdx.x & 31;
    int mtiles = (M + 15) >> 4;
    if (gwave >= mtiles * NGRP) return;          // wave-uniform: EXEC all-1 at WMMA
    int tm = gwave / NGRP, tg = gwave % NGRP;
    int row = tm * 16 + (lane & 15);
    int kh  = (lane >> 4) << 1;                  // 0 or 2

    const float* ap = A + (size_t)row * K + kh;

    int  col0 = tg * TILEW + (lane & 15);
    // clamp only matters when NCOL < 16 (z2 GEMM); keeps loads unconditional
    int  ccol0 = (NCOL < 16 && col0 >= NCOL) ? 0 : col0;
    const float* bbase = B + (size_t)kh * NCOL + ccol0;

    v8f zero = {0.f, 0.f, 0.f, 0.f, 0.f, 0.f, 0.f, 0.f};
    v8f acc[NT];
#pragma unroll
    for (int t = 0; t < NT; ++t) acc[t] = zero;

#pragma unroll
    for (int k = 0; k < K; k += 4) {
        v2f a = *reinterpret_cast<const v2f*>(ap + k);
#pragma unroll
        for (int t = 0; t < NT; ++t) {
            v2f b;
            b.x = bbase[(size_t)k * NCOL + 16 * t];
            b.y = bbase[(size_t)(k + 1) * NCOL + 16 * t];
            if (NCOL < 16 && col0 >= NCOL) { b.x = 0.f; b.y = 0.f; }  // v_cndmask
            acc[t] = __builtin_amdgcn_wmma_f32_16x16x4_f32(
                false, a, false, b, (short)0, acc[t], false, false);
        }
    }

    int rb = tm * 16 + ((lane >> 4) << 3);
#pragma unroll
    for (int t = 0; t < NT; ++t) {
        int col = col0 + 16 * t;
        if (!(NCOL < 16) || col < NCOL) {
            float* cp = C + (size_t)rb * NCOL + col;
#pragma unroll
            for (int v = 0; v < 8; ++v) cp[(size_t)v * NCOL] = acc[t][v];
        }
    }
}

// ---- attention coefficients, layer 1 (H=2, F=64) ----
__global__ void k_coef1(const float* __restrict__ z, const float* __restrict__ al,
                        const float* __restrict__ ar, float* __restrict__ el,
                        float* __restrict__ er) {
    int n = blockIdx.x * blockDim.x + threadIdx.x;
    if (n >= NN) return;
    float e0 = 0.f, e1 = 0.f, r0 = 0.f, r1 = 0.f;
#pragma unroll 8
    for (int f = 0; f < 64; ++f) {
        float z0 = z[n * 128 + f], z1v = z[n * 128 + 64 + f];
        e0 += z0 * al[f];      e1 += z1v * al[64 + f];
        r0 += z0 * ar[f];      r1 += z1v * ar[64 + f];
    }
    el[n * 2] = e0; el[n * 2 + 1] = e1;
    er[n * 2] = r0; er[n * 2 + 1] = r1;
}

__global__ void k_max1(const int* __restrict__ src, const int* __restrict__ dst,
                       const float* __restrict__ el, const float* __restrict__ er,
                       float* __restrict__ m) {
    int g = blockIdx.x * blockDim.x + threadIdx.x;
    if (g >= NE * 2) return;
    int e = g >> 1, h = g & 1;
    int d = dst[e];
    float sc = leaky(el[src[e] * 2 + h] + er[d * 2 + h]);
    atomicMaxF(&m[d * 2 + h], sc);
}

__global__ void k_guard(float* __restrict__ m, int n) {
    int i = blockIdx.x * blockDim.x + threadIdx.x;
    if (i < n && m[i] == -INFINITY) m[i] = 0.f;
}

__global__ void k_sum1(const int* __restrict__ src, const int* __restrict__ dst,
                       const float* __restrict__ el, const float* __restrict__ er,
                       const float* __restrict__ m, float* __restrict__ s) {
    int g = blockIdx.x * blockDim.x + threadIdx.x;
    if (g >= NE * 2) return;
    int e = g >> 1, h = g & 1;
    int d = dst[e];
    float sc = leaky(el[src[e] * 2 + h] + er[d * 2 + h]);
    atomicAdd(&s[d * 2 + h], __expf(sc - m[d * 2 + h]));
}

// ---- conv1 message scatter: one wave32 per edge, 4 f32 columns per lane ----
__global__ void k_msg1(const int* __restrict__ src, const int* __restrict__ dst,
                       const float* __restrict__ el, const float* __restrict__ er,
                       const float* __restrict__ m, const float* __restrict__ s,
                       const float* __restrict__ z, float* __restrict__ out) {
    int gid  = blockIdx.x * blockDim.x + threadIdx.x;
    int edge = gid >> 5;
    int lane = gid & 31;
    if (edge >= NE) return;                      // wave-uniform
    int sN = src[edge], dN = dst[edge];
    int col = lane << 2;                         // 0..124, all 4 cols share one head
    int h = col >> 6;
    float sc    = leaky(el[sN * 2 + h] + er[dN * 2 + h]);
    float alpha = __expf(sc - m[dN * 2 + h]) / fmaxf(s[dN * 2 + h], 1e-9f);
    const float4 zv = *reinterpret_cast<const float4*>(z + (size_t)sN * 128 + col);
    float* o = out + (size_t)dN * 128 + col;
    atomicAdd(o + 0, alpha * zv.x);
    atomicAdd(o + 1, alpha * zv.y);
    atomicAdd(o + 2, alpha * zv.z);
    atomicAdd(o + 3, alpha * zv.w);
}

// ---- bias + relu + faithful permute(1,0,2).reshape(N, H*F) ----
__global__ void k_hcat(const float* __restrict__ out1, const float* __restrict__ b1,
                       float* __restrict__ hcat) {
    int g = blockIdx.x * blockDim.x + threadIdx.x;
    if (g >= NN * 128) return;
    int r = g >> 7, c = g & 127;
    int h  = (r < NN / 2) ? 0 : 1;
    int rr = r - h * (NN / 2);
    int node = 2 * rr + (c >> 6);
    int f = c & 63;
    float v = out1[node * 128 + h * 64 + f] + b1[h * 64 + f];
    hcat[g] = fmaxf(v, 0.f);
}

__global__ void k_addbias(float* __restrict__ x, const float* __restrict__ b,
                          int cols, int n) {
    int i = blockIdx.x * blockDim.x + threadIdx.x;
    if (i < n) x[i] += b[i % cols];
}

// ---- layer 2 attention (H=1, C=8) ----
__global__ void k_coef2(const float* __restrict__ z, const float* __restrict__ al,
                        const float* __restrict__ ar, float* __restrict__ el,
                        float* __restrict__ er) {
    int n = blockIdx.x * blockDim.x + threadIdx.x;
    if (n >= NN) return;
    float a = 0.f, r = 0.f;
#pragma unroll
    for (int c = 0; c < 8; ++c) {
        float zv = z[n * 8 + c];
        a += zv * al[c];
        r += zv * ar[c];
    }
    el[n] = a; er[n] = r;
}

__global__ void k_max2(const int* __restrict__ src, const int* __restrict__ dst,
                       const float* __restrict__ el, const float* __restrict__ er,
                       float* __restrict__ m) {
    int e = blockIdx.x * blockDim.x + threadIdx.x;
    if (e >= NE) return;
    int d = dst[e];
    atomicMaxF(&m[d], leaky(el[src[e]] + er[d]));
}

__global__ void k_sum2(const int* __restrict__ src, const int* __restrict__ dst,
                       const float* __restrict__ el, const float* __restrict__ er,
                       const float* __restrict__ m, float* __restrict__ s) {
    int e = blockIdx.x * blockDim.x + threadIdx.x;
    if (e >= NE) return;
    int d = dst[e];
    float sc = leaky(el[src[e]] + er[d]);
    atomicAdd(&s[d], __expf(sc - m[d]));
}

__global__ void k_msg2(const int* __restrict__ src, const int* __restrict__ dst,
                       const float* __restrict__ el, const float* __restrict__ er,
                       const float* __restrict__ m, const float* __restrict__ s,
                       const float* __restrict__ z, float* __restrict__ out) {
    int e = blockIdx.x * blockDim.x + threadIdx.x;
    if (e >= NE) return;
    int sN = src[e], dN = dst[e];
    float sc    = leaky(el[sN] + er[dN]);
    float alpha = __expf(sc - m[dN]) / fmaxf(s[dN], 1e-9f);
    const float4* zp = reinterpret_cast<const float4*>(z + (size_t)sN * 8);
    float4 z0 = zp[0], z1 = zp[1];
    float* o = out + (size_t)dN * 8;
    atomicAdd(o + 0, alpha * z0.x); atomicAdd(o + 1, alpha * z0.y);
    atomicAdd(o + 2, alpha * z0.z); atomicAdd(o + 3, alpha * z0.w);
    atomicAdd(o + 4, alpha * z1.x); atomicAdd(o + 5, alpha * z1.y);
    atomicAdd(o + 6, alpha * z1.z); atomicAdd(o + 7, alpha * z1.w);
}

template <int K, int NCOL, int NT>
static inline void launch_gemm(const float* A, const float* B, float* C,
                               int M, hipStream_t stream) {
    constexpr int NGRP = (NCOL + 16 * NT - 1) / (16 * NT);
    int waves = ((M + 15) / 16) * NGRP;
    int threads = waves * 32;
    k_gemm_t<K, NCOL, NT><<<(threads + 255) / 256, 256, 0, stream>>>(A, B, C, M);
}

extern "C" void kernel_launch(void* const* d_in, const int* in_sizes, int n_in,
                              void* d_out_v, int out_size, void* d_ws, size_t ws_size,
                              hipStream_t stream) {
    const float* in_feat = (const float*)d_in[0];
    const float* e_feat  = (const float*)d_in[1];
    const int*   src     = (const int*)d_in[2];
    const int*   dst     = (const int*)d_in[3];
    const float* We      = (const float*)d_in[4];
    const float* be      = (const float*)d_in[5];
    const float* W1      = (const float*)d_in[6];
    const float* al1     = (const float*)d_in[7];
    const float* ar1     = (const float*)d_in[8];
    const float* b1      = (const float*)d_in[9];
    const float* Wi      = (const float*)d_in[10];
    const float* bi      = (const float*)d_in[11];
    const float* W2      = (const float*)d_in[12];
    const float* al2     = (const float*)d_in[13];
    const float* ar2     = (const float*)d_in[14];
    const float* b2      = (const float*)d_in[15];
    float* out = (float*)d_out_v;

    // workspace carve-up (floats)
    float* ws = (float*)d_ws;
    size_t o = 0;
    float* n_e   = ws + o; o += NN;
    float* feats = ws + o; o += (size_t)NN * 128;   // reused as hcat after msg1
    float* z1    = ws + o; o += (size_t)NN * 128;   // reused as h2 / z2 after msg1
    float* el1   = ws + o; o += (size_t)NN * 2;
    float* er1   = ws + o; o += (size_t)NN * 2;
    float* m1    = ws + o; o += (size_t)NN * 2;
    float* s1    = ws + o; o += (size_t)NN * 2;
    float* out1  = ws + o; o += (size_t)NN * 128;
    float* el2   = ws + o; o += NN;
    float* er2   = ws + o; o += NN;
    float* m2    = ws + o; o += NN;
    float* s2    = ws + o; o += NN;
    float* hcat  = feats;
    float* h2    = z1;
    float* z2    = z1 + (size_t)NN * 64;

    const int T = 256;

    // init accumulators every call (deterministic, capture-safe)
    hipMemsetAsync(n_e,  0, (size_t)NN * 4, stream);
    hipMemsetAsync(s1,   0, (size_t)NN * 2 * 4, stream);
    hipMemsetAsync(out1, 0, (size_t)NN * 128 * 4, stream);
    hipMemsetAsync(s2,   0, (size_t)NN * 4, stream);
    hipMemsetAsync(out,  0, (size_t)NN * 8 * 4, stream);
    k_fill<<<(2 * NN + T - 1) / T, T, 0, stream>>>(m1, -INFINITY, 2 * NN);
    k_fill<<<(NN + T - 1) / T, T, 0, stream>>>(m2, -INFINITY, NN);

    // edge encoder + scatter, build feats
    k_edge_embed<<<(NE + T - 1) / T, T, 0, stream>>>(e_feat, We, be, dst, n_e);
    k_feats<<<(NN * 128 + T - 1) / T, T, 0, stream>>>(in_feat, n_e, feats);

    // conv1
    launch_gemm<128, 128, 4>(feats, W1, z1, NN, stream);
    k_coef1<<<(NN + T - 1) / T, T, 0, stream>>>(z1, al1, ar1, el1, er1);
    k_max1<<<(2 * NE + T - 1) / T, T, 0, stream>>>(src, dst, el1, er1, m1);
    k_guard<<<(2 * NN + T - 1) / T, T, 0, stream>>>(m1, 2 * NN);
    k_sum1<<<(2 * NE + T - 1) / T, T, 0, stream>>>(src, dst, el1, er1, m1, s1);
    k_msg1<<<(NE * 32 + T - 1) / T, T, 0, stream>>>(src, dst, el1, er1, m1, s1, z1, out1);

    // inter layer: bias+relu+permute-reshape, then Linear
    k_hcat<<<(NN * 128 + T - 1) / T, T, 0, stream>>>(out1, b1, hcat);
    launch_gemm<128, 64, 4>(hcat, Wi, h2, NN, stream);
    k_addbias<<<(NN * 64 + T - 1) / T, T, 0, stream>>>(h2, bi, 64, NN * 64);

    // conv2
    launch_gemm<64, 8, 1>(h2, W2, z2, NN, stream);
    k_coef2<<<(NN + T - 1) / T, T, 0, stream>>>(z2, al2, ar2, el2, er2);
    k_max2<<<(NE + T - 1) / T, T, 0, stream>>>(src, dst, el2, er2, m2);
    k_guard<<<(NN + T - 1) / T, T, 0, stream>>>(m2, NN);
    k_sum2<<<(NE + T - 1) / T, T, 0, stream>>>(src, dst, el2, er2, m2, s2);
    k_msg2<<<(NE + T - 1) / T, T, 0, stream>>>(src, dst, el2, er2, m2, s2, z2, out);
    k_addbias<<<(NN * 8 + T - 1) / T, T, 0, stream>>>(out, b2, 8, NN * 8);
}